// ROIPool_3d_70265664962687
// MI455X (gfx1250) — compile-verified
//
#include <hip/hip_runtime.h>
#include <cstdint>

// ROIPool 3D (adaptive max-pool over ROI boxes), CDNA5 / gfx1250.
// feat: (1, 128, 48, 48, 48) fp32    rois: (1, 32, 7) fp32
// out:  (32, 128, 4, 4, 4) fp32
//
// One 256-thread block (8 wave32) per (roi, 4-channel group). Wave 0 issues a
// single Tensor Data Mover DMA (tensor_load_to_lds) describing a 4D tile
// (Lz x Ly x Lx x 4 channels) of the feature tensor, waits on TENSORcnt,
// then all 8 waves compute the 4x64 output cells from LDS.

#define FD      48                   // spatial dim (D == H == W)
#define FC      128                  // channels
#define NROI    32
#define CH_PER_BLK 4
#define BOXMAX  (17 * 17 * 17)       // max ROI box volume per channel

typedef uint32_t v4u __attribute__((ext_vector_type(4)));
typedef int      v4i __attribute__((ext_vector_type(4)));
typedef int      v8i __attribute__((ext_vector_type(8)));

__global__ __launch_bounds__(256)
void roipool3d_tdm_kernel(const float* __restrict__ feat,
                          const float* __restrict__ rois,
                          float* __restrict__ out) {
    extern __shared__ float box[];       // CH_PER_BLK boxes, packed at N-float stride

    const int blk = blockIdx.x;          // 0 .. NROI*(FC/CH_PER_BLK)-1
    const int r   = blk >> 5;            // roi index      (blk / 32)
    const int cg  = blk & 31;            // channel group  (blk % 32)
    const int c0  = cg * CH_PER_BLK;
    const int tid = threadIdx.x;         // 0 .. 255

    // ---- decode ROI (uniform across block) -------------------------------
    const float* rp = rois + r * 7;      // [batch, x0,y0,z0, x1,y1,z1] (x4 scale)
    const int x0 = (int)rintf(rp[1] * 0.25f);
    const int y0 = (int)rintf(rp[2] * 0.25f);
    const int z0 = (int)rintf(rp[3] * 0.25f);
    const int x1 = (int)rintf(rp[4] * 0.25f + 1.0f);
    const int y1 = (int)rintf(rp[5] * 0.25f + 1.0f);
    const int z1 = (int)rintf(rp[6] * 0.25f + 1.0f);
    const int Lx = x1 - x0, Ly = y1 - y0, Lz = z1 - z0;
    const int N  = Lx * Ly * Lz;         // per-channel box volume

    // ---- wave 0: one TDM descriptor stages all 4 channel boxes -----------
    if (tid < 32) {
        const uint64_t gaddr = (uint64_t)(uintptr_t)
            (feat + ((((size_t)c0 * FD + x0) * FD + y0) * FD + z0));
        const uint32_t lds = (uint32_t)(uintptr_t)(&box[0]);   // LDS byte addr = addr[31:0]

        // D# group 0: count=1 | lds_addr | global_addr[56:0] | type=2
        const v4u g0 = { 1u,
                         lds,
                         (uint32_t)gaddr,
                         (uint32_t)((gaddr >> 32) & 0x01FFFFFFu) | (2u << 30) };

        // D# group 1 (ISA bit map):
        //  d0: workgroup_mask=0 | data_size=2 (4B) at [17:16]
        //  d1: atomic_barrier_addr=0 | tensor_dim0[15:0]<<16
        //  d2: tensor_dim0[31:16]=0 | tensor_dim1[15:0]<<16
        //  d3: tensor_dim1[31:16]=0 | tile_dim0<<16
        //  d4: tile_dim1 | tile_dim2<<16
        //  d5: tensor_dim0_stride[31:0]
        //  d6: tensor_dim0_stride[47:32]=0 | tensor_dim1_stride[15:0]<<16
        //  d7: tensor_dim1_stride[47:16]
        const v8i g1 = { (int)(2u << 16),
                         (int)((uint32_t)FD << 16),
                         (int)((uint32_t)FD << 16),
                         (int)((uint32_t)Lz << 16),
                         (int)((uint32_t)Ly | ((uint32_t)Lx << 16)),
                         (int)FD,
                         (int)((uint32_t)(FD * FD) << 16),
                         0 };

        // D# group 2: tensor_dim2 | tensor_dim3 | tensor_dim2_stride | tile_dim3<<16
        const v4i g2 = { FD,
                         FC,
                         FD * FD * FD,
                         (int)((uint32_t)CH_PER_BLK << 16) };

        // D# group 3: tensor_dim3_stride=0, tensor_dim4=0, tile_dim4=0
        const v4i g3 = { 0, 0, 0, 0 };

#if __has_builtin(__builtin_amdgcn_tensor_load_to_lds)
  #if defined(__clang_major__) && __clang_major__ >= 23
        const v8i g4 = { 0, 0, 0, 0, 0, 0, 0, 0 };   // VADDR4: unused, zero-filled
        __builtin_amdgcn_tensor_load_to_lds(g0, g1, g2, g3, g4, 0);
  #else
        __builtin_amdgcn_tensor_load_to_lds(g0, g1, g2, g3, 0);
  #endif
#else
        // Fallback: inline asm with uniform (readfirstlane-forced) descriptor words.
        v4u a0; v8i a1; v4i a2; v4i a3;
        for (int i = 0; i < 4; ++i) a0[i] = __builtin_amdgcn_readfirstlane(g0[i]);
        for (int i = 0; i < 8; ++i) a1[i] = __builtin_amdgcn_readfirstlane(g1[i]);
        for (int i = 0; i < 4; ++i) a2[i] = __builtin_amdgcn_readfirstlane(g2[i]);
        for (int i = 0; i < 4; ++i) a3[i] = __builtin_amdgcn_readfirstlane(g3[i]);
        asm volatile("tensor_load_to_lds %0, %1, %2, %3"
                     :
                     : "s"(a0), "s"(a1), "s"(a2), "s"(a3)
                     : "memory");
#endif

#if __has_builtin(__builtin_amdgcn_s_wait_tensorcnt)
        __builtin_amdgcn_s_wait_tensorcnt(0);
#else
        asm volatile("s_wait_tensorcnt 0" ::: "memory");
#endif
    }
    __syncthreads();

    // ---- each thread: one (channel, cell) = max over its adaptive bin ----
    const int chl  = tid >> 6;           // 0 .. 3
    const int cell = tid & 63;           // ox*16 + oy*4 + oz
    const int oz = cell & 3;
    const int oy = (cell >> 2) & 3;
    const int ox = cell >> 4;

    // AdaptiveMaxPool bins: [floor(i*L/4), ceil((i+1)*L/4))
    const int bx0 = (ox * Lx) >> 2, bx1 = ((ox + 1) * Lx + 3) >> 2;
    const int by0 = (oy * Ly) >> 2, by1 = ((oy + 1) * Ly + 3) >> 2;
    const int bz0 = (oz * Lz) >> 2, bz1 = ((oz + 1) * Lz + 3) >> 2;

    const float* bb = box + (size_t)chl * N;   // TDM packs channel tiles back-to-back

    float m = -INFINITY;
    for (int dx = bx0; dx < bx1; ++dx) {
        for (int dy = by0; dy < by1; ++dy) {
            const float* row = bb + (dx * Ly + dy) * Lz;
            for (int dz = bz0; dz < bz1; ++dz) {
                m = fmaxf(m, row[dz]);
            }
        }
    }

    const int c = c0 + chl;
    out[((size_t)r * FC + c) * 64 + cell] = m;
}

extern "C" void kernel_launch(void* const* d_in, const int* in_sizes, int n_in,
                              void* d_out, int out_size, void* d_ws, size_t ws_size,
                              hipStream_t stream) {
    const float* feat = (const float*)d_in[0];   // (1,128,48,48,48) fp32
    const float* rois = (const float*)d_in[1];   // (1,32,7) fp32
    float* out = (float*)d_out;                  // (32,128,4,4,4) fp32

    const int blocks = NROI * (FC / CH_PER_BLK);           // 1024
    const size_t shmem = (size_t)CH_PER_BLK * BOXMAX * sizeof(float);  // 78608 B
    roipool3d_tdm_kernel<<<blocks, 256, shmem, stream>>>(feat, rois, out);
}